// LightGCN_79534204387833
// MI455X (gfx1250) — compile-verified
//
#include <hip/hip_runtime.h>

typedef float v4f __attribute__((ext_vector_type(4)));

#define EMBED 64
#define EDGES_PER_WAVE 8
#define WAVES_PER_BLOCK 8
#define EDGES_PER_BLOCK 64          // EDGES_PER_WAVE * WAVES_PER_BLOCK
#define TPB 256

// ---------------------------------------------------------------------------
// init: cur = acc = concat(user_emb, item_emb); nxt = 0.
// One-shot source reads are non-temporal (don't pollute L2); cur stores are RT
// so the scatter's source table is L2-resident; acc stores are NT (streaming).
// ---------------------------------------------------------------------------
__global__ __launch_bounds__(TPB) void lgcn_init(
    const v4f* __restrict__ user, const v4f* __restrict__ item,
    v4f* __restrict__ cur, v4f* __restrict__ acc, v4f* __restrict__ nxt,
    long user_f4, long total_f4)
{
    long i = (long)blockIdx.x * blockDim.x + threadIdx.x;
    long stride = (long)gridDim.x * blockDim.x;
    v4f z = {0.f, 0.f, 0.f, 0.f};
    for (; i < total_f4; i += stride) {
        v4f v = (i < user_f4) ? __builtin_nontemporal_load(&user[i])
                              : __builtin_nontemporal_load(&item[i - user_f4]);
        cur[i] = v;                               // RT: keep table in L2
        __builtin_nontemporal_store(v, &acc[i]);  // accumulator streams
        nxt[i] = z;                               // RT: pre-allocate atomic dest
    }
}

// ---------------------------------------------------------------------------
// scatter: one edge per wave32; lane l owns dims 2l,2l+1 (float2 -> b64, the
// 32 lanes cover one full 256-B row). Edge triples are staged into LDS with
// CDNA5 async-to-LDS loads (ASYNCcnt path, NT hint: edge lists stream), then
// broadcast per wave via ds_load. Row update = 2x hardware global_atomic_add_f32
// (no-return form, resolves in L2 where the 128 MB table lives).
// ---------------------------------------------------------------------------
__global__ __launch_bounds__(TPB) void lgcn_scatter(
    const float* __restrict__ cur, float* __restrict__ nxt,
    const int* __restrict__ erow, const int* __restrict__ ecol,
    const float* __restrict__ eval, int nnz)
{
    __shared__ int   s_row[EDGES_PER_BLOCK];
    __shared__ int   s_col[EDGES_PER_BLOCK];
    __shared__ float s_val[EDGES_PER_BLOCK];

    const int base = blockIdx.x * EDGES_PER_BLOCK;
    const int t    = threadIdx.x;

    // Async staging: waves 0-1 -> rows, 2-3 -> cols, 4-5 -> vals (wave-uniform
    // selection, no divergence). Clamped tail addresses are never consumed.
    if (t < 3 * EDGES_PER_BLOCK) {
        const int which = t >> 6;
        const int j     = t & 63;
        int e = base + j;
        if (e >= nnz) e = nnz - 1;
        const void* gsrc;
        const void* ldst;
        if (which == 0)      { gsrc = (const void*)(erow + e); ldst = (const void*)&s_row[j]; }
        else if (which == 1) { gsrc = (const void*)(ecol + e); ldst = (const void*)&s_col[j]; }
        else                 { gsrc = (const void*)(eval + e); ldst = (const void*)&s_val[j]; }
        // low 32 bits of a generic shared pointer == LDS byte offset
        unsigned lds32 = (unsigned)(unsigned long long)ldst;
        asm volatile("global_load_async_to_lds_b32 %0, %1, off th:TH_LOAD_NT"
                     :: "v"(lds32), "v"(gsrc) : "memory");
    }
    asm volatile("s_wait_asynccnt 0" ::: "memory");
    __syncthreads();

    const int wave  = t >> 5;        // wave32
    const int lane  = t & 31;
    const int first = wave * EDGES_PER_WAVE;

    if (base + first < nnz) {
        __builtin_prefetch(cur + (size_t)s_col[first] * EMBED + lane * 2, 0, 1);
    }

    for (int i = 0; i < EDGES_PER_WAVE; ++i) {
        const int idx = first + i;                 // wave-uniform
        if (base + idx >= nnz) break;              // uniform branch

        if (i + 1 < EDGES_PER_WAVE && base + idx + 1 < nnz) {
            __builtin_prefetch(cur + (size_t)s_col[idx + 1] * EMBED + lane * 2, 0, 1);
        }

        const int   row = s_row[idx];
        const int   col = s_col[idx];
        const float val = s_val[idx];

        const float2 src = *(const float2*)(cur + (size_t)col * EMBED + lane * 2);
        const float  mx  = src.x * val;
        const float  my  = src.y * val;

        float* dst = nxt + (size_t)row * EMBED + lane * 2;
        asm volatile("global_atomic_add_f32 %0, %1, off"
                     :: "v"(dst), "v"(mx) : "memory");
        asm volatile("global_atomic_add_f32 %0, %1, off offset:4"
                     :: "v"(dst), "v"(my) : "memory");
    }
}

// ---------------------------------------------------------------------------
// acc += nxt, and zero `to_zero` (the buffer that becomes the next layer's
// scatter destination) in the same streaming pass — saves a 128 MB pass and
// pre-allocates the atomic destination lines in L2.
// nxt stays RT (it becomes the next layer's gather source table).
// ---------------------------------------------------------------------------
__global__ __launch_bounds__(TPB) void lgcn_acc_zero(
    v4f* __restrict__ acc, const v4f* __restrict__ nxt,
    v4f* __restrict__ to_zero, long n4)
{
    long i = (long)blockIdx.x * blockDim.x + threadIdx.x;
    long stride = (long)gridDim.x * blockDim.x;
    v4f z = {0.f, 0.f, 0.f, 0.f};
    for (; i < n4; i += stride) {
        v4f a = __builtin_nontemporal_load(&acc[i]);
        v4f b = nxt[i];
        __builtin_nontemporal_store(a + b, &acc[i]);
        to_zero[i] = z;
    }
}

// final layer: acc = (acc + nxt) * 0.25; everything is last-use -> all NT
__global__ __launch_bounds__(TPB) void lgcn_acc_scale(
    v4f* __restrict__ acc, const v4f* __restrict__ nxt, long n4, float s)
{
    long i = (long)blockIdx.x * blockDim.x + threadIdx.x;
    long stride = (long)gridDim.x * blockDim.x;
    for (; i < n4; i += stride) {
        v4f a = __builtin_nontemporal_load(&acc[i]);
        v4f b = __builtin_nontemporal_load(&nxt[i]);
        __builtin_nontemporal_store((a + b) * s, &acc[i]);
    }
}

// ---------------------------------------------------------------------------
extern "C" void kernel_launch(void* const* d_in, const int* in_sizes, int n_in,
                              void* d_out, int out_size, void* d_ws, size_t ws_size,
                              hipStream_t stream)
{
    const float* user  = (const float*)d_in[0];
    const float* item  = (const float*)d_in[1];
    const int*   erow  = (const int*)d_in[2];
    const int*   ecol  = (const int*)d_in[3];
    const float* evalv = (const float*)d_in[4];
    float* out = (float*)d_out;

    const long user_elems = (long)in_sizes[0];        // NUM_USERS * 64
    const long item_elems = (long)in_sizes[1];        // NUM_ITEMS * 64
    const int  nnz        = in_sizes[2];
    const long total      = user_elems + item_elems;  // N_NODES * 64 floats
    const long total_f4   = total / 4;
    const long user_f4    = user_elems / 4;

    // ping-pong embedding tables in workspace (2 x 128 MB)
    float* cur = (float*)d_ws;
    float* nxt = (float*)((char*)d_ws + (size_t)total * sizeof(float));

    const int ew_blocks   = (int)((total_f4 + TPB - 1) / TPB);
    const int scat_blocks = (nnz + EDGES_PER_BLOCK - 1) / EDGES_PER_BLOCK;

    lgcn_init<<<ew_blocks, TPB, 0, stream>>>(
        (const v4f*)user, (const v4f*)item,
        (v4f*)cur, (v4f*)out, (v4f*)nxt, user_f4, total_f4);

    for (int l = 0; l < 3; ++l) {
        lgcn_scatter<<<scat_blocks, TPB, 0, stream>>>(cur, nxt, erow, ecol, evalv, nnz);
        if (l < 2) {
            // acc += nxt; zero cur (it becomes next layer's atomic destination)
            lgcn_acc_zero<<<ew_blocks, TPB, 0, stream>>>(
                (v4f*)out, (const v4f*)nxt, (v4f*)cur, total_f4);
        } else {
            lgcn_acc_scale<<<ew_blocks, TPB, 0, stream>>>(
                (v4f*)out, (const v4f*)nxt, total_f4, 0.25f);
        }
        float* tmp = cur; cur = nxt; nxt = tmp;
    }
}